// COMA_19267223290289
// MI455X (gfx1250) — compile-verified
//
#include <hip/hip_runtime.h>

// COMA losses: policy_loss, q_value_loss, entropy_loss
// T=256, B=128, A=8, N=64. Memory-bound (~140MB @ 23.3TB/s ~ 6-7us).

#define GAMMA 0.99f
#define LAMBDA 0.95f

constexpr int T = 256, B = 128, A = 8, N = 64;
constexpr int R  = T * B * A;          // 262144 rows of width N
constexpr int BA = B * A;              // 1024
constexpr int ROWS_PER_BLOCK = 8;      // 8 wave32 per 256-thread block
constexpr int NBLK1 = R / ROWS_PER_BLOCK;  // 32768 partials

typedef __attribute__((ext_vector_type(2))) float v2f;
typedef __attribute__((ext_vector_type(8))) float v8f;

// ---------------------------------------------------------------------------
// Kernel 1: per-row softmax stats + gathers + policy/entropy partial sums.
// One wave32 per row; lane l holds columns {2l, 2l+1} (coalesced b64 loads).
// ---------------------------------------------------------------------------
__global__ __launch_bounds__(256) void coma_rowpass(
    const float* __restrict__ logit, const int* __restrict__ action,
    const float* __restrict__ qval,  const float* __restrict__ tqval,
    const float* __restrict__ weight,
    float* __restrict__ qtaken, float* __restrict__ tqtaken,
    float* __restrict__ pol_partial, float* __restrict__ ent_partial)
{
    const int tid  = threadIdx.x;
    const int wid  = tid >> 5;
    const int lane = tid & 31;
    const int row  = blockIdx.x * ROWS_PER_BLOCK + wid;

    const size_t base = (size_t)row * N;
    v2f lg = *(const v2f*)(logit + base + 2 * lane);
    v2f qv = *(const v2f*)(qval  + base + 2 * lane);

    // row max
    float m = fmaxf(lg.x, lg.y);
#pragma unroll
    for (int d = 16; d > 0; d >>= 1) m = fmaxf(m, __shfl_xor(m, d, 32));

    // sum of exp
    float e0 = expf(lg.x - m), e1 = expf(lg.y - m);
    float s = e0 + e1;
#pragma unroll
    for (int d = 16; d > 0; d >>= 1) s += __shfl_xor(s, d, 32);

    const float logZ = m + logf(s);
    const float inv  = 1.0f / s;
    const float p0 = e0 * inv, p1 = e1 * inv;
    const float lp0 = lg.x - logZ, lp1 = lg.y - logZ;

    // baseline = sum(prob*q) ; plogp = sum(prob*logp)
    float bl = p0 * qv.x + p1 * qv.y;
    float en = p0 * lp0 + p1 * lp1;
#pragma unroll
    for (int d = 16; d > 0; d >>= 1) bl += __shfl_xor(bl, d, 32);
#pragma unroll
    for (int d = 16; d > 0; d >>= 1) en += __shfl_xor(en, d, 32);

    const int ai = action[row];                 // 0..63, wave-uniform
    const float cq = (ai & 1) ? qv.y : qv.x;
    const float cl = (ai & 1) ? lp1  : lp0;
    const float q_t  = __shfl(cq, ai >> 1, 32); // q_taken
    const float lp_t = __shfl(cl, ai >> 1, 32); // logp_taken

    const float wt  = weight[row];
    const float adv = q_t - bl;                 // stop_gradient irrelevant (fwd only)
    const float pol = -(lp_t * adv * wt);       // policy_loss term
    const float ent = (-en) * wt;               // entropy term

    __shared__ float sp[ROWS_PER_BLOCK];
    __shared__ float se[ROWS_PER_BLOCK];
    if (lane == 0) {
        qtaken[row]  = q_t;
        tqtaken[row] = tqval[base + ai];        // gather target_q_taken
        sp[wid] = pol;
        se[wid] = ent;
    }
    __syncthreads();
    if (tid == 0) {
        float ap = 0.f, ae = 0.f;
#pragma unroll
        for (int i = 0; i < ROWS_PER_BLOCK; ++i) { ap += sp[i]; ae += se[i]; }
        pol_partial[blockIdx.x] = ap;
        ent_partial[blockIdx.x] = ae;
    }
}

// ---------------------------------------------------------------------------
// Kernel 2: backward lambda-return scan over T per (b,a) column + sq-err sum.
// 1024 threads total (4 blocks x 256); loads coalesced across j.
// ---------------------------------------------------------------------------
__global__ __launch_bounds__(256) void coma_scanpass(
    const float* __restrict__ qtaken, const float* __restrict__ tqtaken,
    const float* __restrict__ reward, const float* __restrict__ weight,
    float* __restrict__ qloss_partial)
{
    const int j = blockIdx.x * 256 + threadIdx.x;   // 0..1023  (b*A + a)
    const int b = j >> 3;

    // t = T-2 init
    float ret = reward[(T - 2) * B + b] + GAMMA * tqtaken[(size_t)(T - 1) * BA + j];
    float d   = ret - qtaken[(size_t)(T - 2) * BA + j];
    float acc = d * d * weight[(size_t)(T - 2) * BA + j];

    for (int t = T - 3; t >= 0; --t) {
        ret = reward[t * B + b] +
              GAMMA * (LAMBDA * ret + (1.0f - LAMBDA) * tqtaken[(size_t)(t + 1) * BA + j]);
        float dd = ret - qtaken[(size_t)t * BA + j];
        acc += dd * dd * weight[(size_t)t * BA + j];
    }

    __shared__ float sm[256];
    sm[threadIdx.x] = acc;
    __syncthreads();
#pragma unroll
    for (int s = 128; s > 0; s >>= 1) {
        if (threadIdx.x < s) sm[threadIdx.x] += sm[threadIdx.x + s];
        __syncthreads();
    }
    if (threadIdx.x == 0) qloss_partial[blockIdx.x] = sm[0];
}

// ---------------------------------------------------------------------------
// Kernel 3: deterministic final reduction using V_WMMA_F32_16X16X4_F32.
// B = all-ones => D[m,n] = sum_k A[m,k] + C[m,n]; chaining C over 512 tiles
// reduces 32768 f32 exactly, in a fixed tree order (deterministic).
// A-tile layout (16x4 f32): lanes 0-15 hold {K=0,1} of row M=lane,
// lanes 16-31 hold {K=2,3} of row M=lane-16.
// ---------------------------------------------------------------------------
__device__ float wmma_sum_64multiple(const float* __restrict__ src, int count, int lane)
{
    v8f c = {};
    v2f ones; ones.x = 1.0f; ones.y = 1.0f;
    const int mrow = lane & 15;
    const int kb   = (lane >> 4) << 1;
    for (int i = 0; i < count; i += 64) {
        v2f a;
        a.x = src[i + mrow * 4 + kb + 0];
        a.y = src[i + mrow * 4 + kb + 1];
        c = __builtin_amdgcn_wmma_f32_16x16x4_f32(
            /*neg_a=*/false, a, /*neg_b=*/false, ones,
            /*c_mod=*/(short)0, c, /*reuse_a=*/false, /*reuse_b=*/false);
    }
    // D column n = lane&15 holds rows M=0..7 in c[0..7] (lanes 0-15) and
    // M=8..15 (lanes 16-31). Sum components then fold lane halves.
    float col = 0.f;
#pragma unroll
    for (int i = 0; i < 8; ++i) col += c[i];
    col += __shfl_xor(col, 16, 32);   // full column sum; identical for all n
    return col;
}

__global__ __launch_bounds__(32) void coma_finalize(
    const float* __restrict__ pol_partial, const float* __restrict__ ent_partial,
    const float* __restrict__ qloss_partial, float* __restrict__ out)
{
    const int lane = threadIdx.x;
    const float polsum = wmma_sum_64multiple(pol_partial, NBLK1, lane);
    const float entsum = wmma_sum_64multiple(ent_partial, NBLK1, lane);
    if (lane == 0) {
        float ql = qloss_partial[0] + qloss_partial[1] +
                   qloss_partial[2] + qloss_partial[3];
        out[0] = polsum / (float)R;                 // policy_loss
        out[1] = ql / (float)((T - 1) * BA);        // q_value_loss
        out[2] = entsum / (float)R;                 // entropy_loss
    }
}

// ---------------------------------------------------------------------------
extern "C" void kernel_launch(void* const* d_in, const int* in_sizes, int n_in,
                              void* d_out, int out_size, void* d_ws, size_t ws_size,
                              hipStream_t stream)
{
    const float* logit  = (const float*)d_in[0];
    const int*   action = (const int*)  d_in[1];
    const float* qval   = (const float*)d_in[2];
    const float* tqval  = (const float*)d_in[3];
    const float* reward = (const float*)d_in[4];
    const float* weight = (const float*)d_in[5];
    float* out = (float*)d_out;

    float* ws          = (float*)d_ws;
    float* qtaken      = ws;                       // R floats
    float* tqtaken     = ws + R;                   // R floats
    float* pol_partial = ws + 2 * R;               // NBLK1 floats
    float* ent_partial = ws + 2 * R + NBLK1;       // NBLK1 floats
    float* ql_partial  = ws + 2 * R + 2 * NBLK1;   // 4 floats

    coma_rowpass<<<NBLK1, 256, 0, stream>>>(logit, action, qval, tqval, weight,
                                            qtaken, tqtaken, pol_partial, ent_partial);
    coma_scanpass<<<BA / 256, 256, 0, stream>>>(qtaken, tqtaken, reward, weight,
                                                ql_partial);
    coma_finalize<<<1, 32, 0, stream>>>(pol_partial, ent_partial, ql_partial, out);
}